// PatchManifold_41472204210223
// MI455X (gfx1250) — compile-verified
//
#include <hip/hip_runtime.h>

typedef float v2f __attribute__((ext_vector_type(2)));
typedef float v8f __attribute__((ext_vector_type(8)));

#define H_IN   2560
#define W_IN   2560
#define CHANS  24          // B*C = 8*3
#define PHH    64          // patch height (h)
#define PWW    64          // patch width  (w)
#define TILE_DH 8          // hh rows per block
#define NBLK   (CHANS * (PHH / TILE_DH))   // 192 blocks
#define PITCH  517         // 8*64 + 5 (odd dword stride per patch -> conflict-free LDS)
#define D_TOT  (CHANS * PHH * PWW)         // 98304

// ---------------------------------------------------------------------------
// Kernel A: fused bilinear resize (exact 0.5 fracs) + per-block pair partials.
// Each block: channel ch, hh-tile hb. Produces a 16x16 partial sum matrix.
// ---------------------------------------------------------------------------
__global__ __launch_bounds__(256)
void pm_partial_kernel(const float* __restrict__ x, float* __restrict__ part)
{
    __shared__ float lds[16 * PITCH];   // 16 patches x (8x64) d-tile, odd pitch
    const int blk = blockIdx.x;
    const int ch  = blk >> 3;           // 0..23
    const int hb  = blk & 7;            // 0..7
    const int t   = threadIdx.x;        // 0..255

    const size_t chbase = (size_t)ch * H_IN * W_IN;

    // Cooperative fused resize into LDS: y[orow][ocol] =
    //   (x[r0][c0]*.5 + x[r1][c0]*.5)*.5 + (x[r0][c1]*.5 + x[r1][c1]*.5)*.5
    // with r0=10*orow+4, c0=10*ocol+4 (frac is exactly 0.5 at 10x downscale).
    for (int e = t; e < 16 * TILE_DH * PWW; e += 256) {
        const int a  = e >> 9;          // patch id 0..15 (a = n*4 + m)
        const int r  = e & 511;
        const int dh = r >> 6;          // 0..7
        const int ww = r & 63;          // 0..63
        const int na = a >> 2, ma = a & 3;
        const int orow = na * PHH + hb * TILE_DH + dh;   // 0..255
        const int ocol = ma * PWW + ww;                  // 0..255
        const int r0 = 10 * orow + 4;
        const int c0 = 10 * ocol + 4;                    // even -> 8B aligned
        const float* p0 = x + chbase + (size_t)r0 * W_IN + c0;
        const float* p1 = p0 + W_IN;
        const v2f u = *(const v2f*)p0;                   // cols c0, c0+1 of row r0
        const v2f v = *(const v2f*)p1;                   // cols c0, c0+1 of row r0+1
        const float t0 = u.x * 0.5f + v.x * 0.5f;        // row pass (match ref order)
        const float t1 = u.y * 0.5f + v.y * 0.5f;
        lds[a * PITCH + dh * 64 + ww] = t0 * 0.5f + t1 * 0.5f;  // col pass
    }
    __syncthreads();

    // Pair-per-thread: thread t owns pair (a,b). LDS b-gather is bank-conflict
    // free because PITCH is odd (517 mod 64 = 5, order-64 generator).
    const int a = t >> 4, b = t & 15;
    const float* pa = lds + a * PITCH;
    const float* pb = lds + b * PITCH;
    float acc = 0.0f;
#pragma unroll 4
    for (int i = 0; i < TILE_DH * PWW; ++i) {
        const float d = pa[i] - pb[i];
        acc += __expf(-d * d);
    }
    part[blk * 256 + t] = acc;   // partial matrix stored row-major [a][b]
}

// ---------------------------------------------------------------------------
// Kernel B: exact reduction of 192 partial 16x16 matrices via f32 WMMA with
// stacked-identity B (D = sum_p M_p), then mean + stable 2-smallest masking.
// Single wave32, EXEC all-1s through the WMMA loop.
// ---------------------------------------------------------------------------
__global__ __launch_bounds__(32)
void pm_combine_kernel(const float* __restrict__ part, float* __restrict__ out)
{
    __shared__ float aff[256];
    const int L     = threadIdx.x;        // 0..31
    const int col   = L & 15;             // N for B/C layout
    const int khalf = (L < 16) ? 0 : 2;   // A/B: lanes 0-15 hold K0/K1, 16-31 hold K2/K3
    const int row   = (L < 16) ? L : (L - 16);

    // 4 identity-slice B chunks: B_q[k][n] = ((4q + k) == n)
    v2f bmat[4];
#pragma unroll
    for (int q = 0; q < 4; ++q) {
        bmat[q].x = (4 * q + khalf     == col) ? 1.0f : 0.0f;
        bmat[q].y = (4 * q + khalf + 1 == col) ? 1.0f : 0.0f;
    }

    v8f c = {};  // C/D accumulator, 16x16 f32 WMMA layout
    for (int p = 0; p < NBLK; ++p) {
        const float* pp = part + p * 256 + row * 16 + khalf;
#pragma unroll
        for (int q = 0; q < 4; ++q) {
            v2f amat;
            amat.x = pp[4 * q];          // A[m][k]=M_p[m][4q+k]
            amat.y = pp[4 * q + 1];
            // D = A x B_q + C  (B_q identity slice => accumulates M_p into C)
            c = __builtin_amdgcn_wmma_f32_16x16x4_f32(
                    false, amat, false, bmat[q], (short)0, c, false, false);
        }
    }

    // Unpack C layout -> LDS: vgpr r is row m=r (lanes 0-15) / m=r+8 (lanes 16-31)
    const float invD = 1.0f / (float)D_TOT;
#pragma unroll
    for (int r = 0; r < 8; ++r) {
        const int m = (L < 16) ? r : (r + 8);
        aff[m * 16 + col] = c[r] * invD;
    }
    __syncthreads();

    // Stable argsort semantics: zero the 2 smallest per row, lowest index on ties.
    if (L < 16) {
        float v[16];
#pragma unroll
        for (int j = 0; j < 16; ++j) v[j] = aff[L * 16 + j];
        int i1 = 0; float m1 = v[0];
        for (int j = 1; j < 16; ++j) if (v[j] < m1) { m1 = v[j]; i1 = j; }
        int i2 = -1; float m2 = 3.4e38f;
        for (int j = 0; j < 16; ++j) { if (j == i1) continue; if (v[j] < m2) { m2 = v[j]; i2 = j; } }
        v[i1] = 0.0f;
        v[i2] = 0.0f;
        for (int j = 0; j < 16; ++j) out[L * 16 + j] = v[j];
    }
}

extern "C" void kernel_launch(void* const* d_in, const int* in_sizes, int n_in,
                              void* d_out, int out_size, void* d_ws, size_t ws_size,
                              hipStream_t stream) {
    (void)in_sizes; (void)n_in; (void)out_size; (void)ws_size;
    const float* targets = (const float*)d_in[0];
    float* part = (float*)d_ws;                 // 192 * 256 floats = 192 KB scratch
    float* out  = (float*)d_out;                // 16x16 fp32

    pm_partial_kernel<<<NBLK, 256, 0, stream>>>(targets, part);
    pm_combine_kernel<<<1, 32, 0, stream>>>(part, out);
}